// DeeperGCN_85899345920722
// MI455X (gfx1250) — compile-verified
//
#include <hip/hip_runtime.h>
#include <math.h>

typedef float v2f __attribute__((ext_vector_type(2)));
typedef float v8f __attribute__((ext_vector_type(8)));

#define N_NODES   30000
#define N_EDGES   480000
#define NUM_FEATS 1433
#define NUM_CLS   7
#define LN_EPS    1e-5f
#define GEN_EPS   1e-7f

// ---------- order-preserving float<->uint encoding for atomic max ----------
__device__ __forceinline__ unsigned fenc(float f) {
  unsigned u = __float_as_uint(f);
  return (u & 0x80000000u) ? ~u : (u | 0x80000000u);
}
__device__ __forceinline__ float fdec(unsigned u) {
  return __uint_as_float((u & 0x80000000u) ? (u ^ 0x80000000u) : ~u);
}
// fenc(-inf) == 0x007FFFFF
#define ENC_NEG_INF 0x007FFFFFu

// ---------- scratch init: m_enc=-inf, denom=0, agg=0 ----------
__global__ void k_conv_init(unsigned* __restrict__ m_enc, float* __restrict__ denom,
                            float* __restrict__ agg, int n) {
  int i = blockIdx.x * blockDim.x + threadIdx.x;
  if (i < n) { m_enc[i] = ENC_NEG_INF; denom[i] = 0.f; agg[i] = 0.f; }
}

// ---------- pass 1: per-(edge,channel) segment max of logits ----------
__global__ void k_edge_max(const int* __restrict__ src, const int* __restrict__ dst,
                           const float* __restrict__ xin, int ldx,
                           unsigned* __restrict__ m_enc, const float* __restrict__ tptr,
                           int dshift, int total) {
  int i = blockIdx.x * blockDim.x + threadIdx.x;
  if (i >= total) return;
  int d = 1 << dshift;
  int e = i >> dshift, c = i & (d - 1);
  float t = tptr[0];
  float msg = fmaxf(xin[src[e] * ldx + c], 0.f) + GEN_EPS;
  atomicMax(&m_enc[dst[e] * d + c], fenc(msg * t));
}

// ---------- pass 2: alpha = exp(logit - m), accumulate denom and alpha*msg ----------
__global__ void k_edge_sum(const int* __restrict__ src, const int* __restrict__ dst,
                           const float* __restrict__ xin, int ldx,
                           const unsigned* __restrict__ m_enc,
                           float* __restrict__ denom, float* __restrict__ agg,
                           const float* __restrict__ tptr, int dshift, int total) {
  int i = blockIdx.x * blockDim.x + threadIdx.x;
  if (i >= total) return;
  int d = 1 << dshift;
  int e = i >> dshift, c = i & (d - 1);
  float t = tptr[0];
  int dn = dst[e];
  float msg = fmaxf(xin[src[e] * ldx + c], 0.f) + GEN_EPS;
  float m = fdec(m_enc[dn * d + c]);
  if (m == -__builtin_inff()) m = 0.f;   // empty segment -> 0 (reference isfinite guard)
  float a = expf(msg * t - m);
  atomicAdd(&denom[dn * d + c], a);
  atomicAdd(&agg[dn * d + c], a * msg);
}

// ---------- agg = agg/max(denom,1e-16) + x   (in place in agg) ----------
__global__ void k_combine(const float* __restrict__ denom, float* __restrict__ agg,
                          const float* __restrict__ xin, int ldx, int dshift, int total) {
  int i = blockIdx.x * blockDim.x + threadIdx.x;
  if (i >= total) return;
  int d = 1 << dshift;
  int n = i >> dshift, c = i & (d - 1);
  agg[i] = agg[i] / fmaxf(denom[i], 1e-16f) + xin[n * ldx + c];
}

// ---------- fp32 WMMA GEMM: C[M x Ncols] = A[M x K] * B[K x Ncols] + bias ----------
// One wave -> one 16x16 tile via V_WMMA_F32_16X16X4_F32.
// Main K loop is branch-free (no per-load EXEC guards); ragged K (encoder's 1433)
// handled by a single tail step with clamped addresses + v_cndmask value masking.
__global__ void k_gemm_wmma(const float* __restrict__ A, int lda,
                            const float* __restrict__ B, int ldb,
                            const float* __restrict__ bias,
                            float* __restrict__ C, int ldc,
                            int M, int K, int Ncols) {
  int MT = M >> 4;
  int tiles = MT * (Ncols >> 4);
  int wave = blockIdx.x * (blockDim.x >> 5) + (threadIdx.x >> 5);
  if (wave >= tiles) return;           // wave-uniform: EXEC all-ones for WMMA
  int mt = wave % MT, nt = wave / MT;  // consecutive waves share the B tile (cache reuse)
  int lane = threadIdx.x & 31;
  int half = lane >> 4, lm = lane & 15;
  int ks0 = half << 1;                 // lanes 0-15: K=kk,kk+1 ; lanes 16-31: K=kk+2,kk+3

  const float* ap = A + ((mt << 4) + lm) * lda + ks0;  // this lane's A row, k cursor
  const float* bp = B + (nt << 4) + lm + ks0 * ldb;    // this lane's B col, k cursor
  int bstep = 4 * ldb;

  v8f acc = {};
  int Kmain = K & ~3;
#pragma unroll 2
  for (int kk = 0; kk < Kmain; kk += 4) {
    v2f a, b;
    a[0] = ap[0];
    a[1] = ap[1];
    b[0] = bp[0];
    b[1] = bp[ldb];
    acc = __builtin_amdgcn_wmma_f32_16x16x4_f32(false, a, false, b, (short)0, acc,
                                                false, false);
    ap += 4;
    bp += bstep;
  }
  if (Kmain < K) {                     // ragged tail, uniform branch
    int ks = Kmain + ks0;
    int kc0 = (ks     < K) ? ks0     : (K - 1 - Kmain);  // clamped, in-bounds offsets
    int kc1 = (ks + 1 < K) ? ks0 + 1 : (K - 1 - Kmain);
    float a0 = ap[kc0 - ks0], a1 = ap[kc1 - ks0];
    float b0 = bp[(kc0 - ks0) * ldb], b1 = bp[(kc1 - ks0) * ldb];
    v2f a, b;
    a[0] = (ks     < K) ? a0 : 0.f;    // value select -> v_cndmask, no EXEC branch
    a[1] = (ks + 1 < K) ? a1 : 0.f;
    b[0] = (ks     < K) ? b0 : 0.f;
    b[1] = (ks + 1 < K) ? b1 : 0.f;
    acc = __builtin_amdgcn_wmma_f32_16x16x4_f32(false, a, false, b, (short)0, acc,
                                                false, false);
  }
  int col = (nt << 4) + lm;
  float bv = bias ? bias[col] : 0.f;
#pragma unroll
  for (int r = 0; r < 8; ++r) {
    // C/D layout: VGPR r -> M = r + 8*half, N = lm
    C[((mt << 4) + r + (half << 3)) * ldc + col] = acc[r] + bv;
  }
}

// ---------- row LayerNorm (+optional ReLU), arbitrary in/out strides ----------
__global__ void k_row_ln(const float* __restrict__ in, int ldin,
                         float* __restrict__ out, int ldout,
                         const float* __restrict__ g, const float* __restrict__ b,
                         int w, int doRelu) {
  __shared__ float s0[128], s1[128];
  const float* x = in + (long)blockIdx.x * ldin;
  float sum = 0.f, sq = 0.f;
  for (int c = threadIdx.x; c < w; c += blockDim.x) {
    float v = x[c]; sum += v; sq += v * v;
  }
  s0[threadIdx.x] = sum; s1[threadIdx.x] = sq;
  __syncthreads();
  for (int off = blockDim.x >> 1; off > 0; off >>= 1) {
    if ((int)threadIdx.x < off) {
      s0[threadIdx.x] += s0[threadIdx.x + off];
      s1[threadIdx.x] += s1[threadIdx.x + off];
    }
    __syncthreads();
  }
  float mu  = s0[0] / w;
  float var = s1[0] / w - mu * mu;
  float inv = rsqrtf(var + LN_EPS);
  float* y = out + (long)blockIdx.x * ldout;
  for (int c = threadIdx.x; c < w; c += blockDim.x) {
    float v = (x[c] - mu) * inv * g[c] + b[c];
    if (doRelu) v = fmaxf(v, 0.f);
    y[c] = v;
  }
}

// ---------- logits: [N x 512] @ [512 x 7] + bias ----------
__global__ void k_logits(const float* __restrict__ xx, const float* __restrict__ W,
                         const float* __restrict__ bias, float* __restrict__ outL) {
  int i = blockIdx.x * blockDim.x + threadIdx.x;
  if (i >= N_NODES * NUM_CLS) return;
  int n = i / NUM_CLS, j = i % NUM_CLS;
  const float* xr = xx + (long)n * 512;
  float s = bias[j];
  for (int k = 0; k < 512; ++k) s += xr[k] * W[k * NUM_CLS + j];
  outL[i] = s;
}

// =================== host-side orchestration ===================
static inline void launch_gemm(const float* A, int lda, const float* B, int ldb,
                               const float* bias, float* C, int ldc,
                               int M, int K, int Ncols, hipStream_t s) {
  int tiles = (M >> 4) * (Ncols >> 4);
  int blocks = (tiles + 7) / 8;          // 8 waves (256 threads) per block
  k_gemm_wmma<<<blocks, 256, 0, s>>>(A, lda, B, ldb, bias, C, ldc, M, K, Ncols);
}

static void run_gen_conv(const float* xin, int ldx, int d, int dshift,
                         const float* tptr, const float* W1, const float* b1,
                         const float* lng, const float* lnb,
                         const float* W2, const float* b2,
                         float* outC, int ldout,
                         const int* src, const int* dst,
                         unsigned* m_enc, float* denom, float* agg, float* T1,
                         hipStream_t s) {
  int nd  = N_NODES * d;
  int tot = N_EDGES * d;                 // <= 122.88M, fits int
  k_conv_init<<<(nd + 255) / 256, 256, 0, s>>>(m_enc, denom, agg, nd);
  k_edge_max <<<(tot + 255) / 256, 256, 0, s>>>(src, dst, xin, ldx, m_enc, tptr, dshift, tot);
  k_edge_sum <<<(tot + 255) / 256, 256, 0, s>>>(src, dst, xin, ldx, m_enc, denom, agg, tptr,
                                                dshift, tot);
  k_combine  <<<(nd + 255) / 256, 256, 0, s>>>(denom, agg, xin, ldx, dshift, nd);
  int n1 = 2 * d;
  // h @ W1 + b1 -> T1 [N x 2d]
  launch_gemm(agg, d, W1, n1, b1, T1, n1, N_NODES, d, n1, s);
  // LayerNorm + ReLU in place on T1
  k_row_ln<<<N_NODES, 128, 0, s>>>(T1, n1, T1, n1, lng, lnb, n1, 1);
  // T1 @ W2 + b2 -> outC [N x d] (gen_conv output)
  launch_gemm(T1, n1, W2, d, b2, outC, ldout, N_NODES, n1, d, s);
}

extern "C" void kernel_launch(void* const* d_in, const int* in_sizes, int n_in,
                              void* d_out, int out_size, void* d_ws, size_t ws_size,
                              hipStream_t stream) {
  (void)in_sizes; (void)n_in; (void)out_size; (void)ws_size;
  const float* x     = (const float*)d_in[0];
  const int*   ei    = (const int*)d_in[1];
  const int*   src   = ei;
  const int*   dst   = ei + N_EDGES;
  const float* enc_W = (const float*)d_in[2];
  const float* enc_b = (const float*)d_in[3];

  const float *Lt[4], *LW1[4], *Lb1[4], *Llng[4], *Llnb[4], *LW2[4], *Lb2[4], *Lng[4], *Lnb[4];
  for (int i = 0; i < 4; ++i) {
    int b0 = 4 + 9 * i;
    Lt[i]   = (const float*)d_in[b0 + 0];
    LW1[i]  = (const float*)d_in[b0 + 1];
    Lb1[i]  = (const float*)d_in[b0 + 2];
    Llng[i] = (const float*)d_in[b0 + 3];
    Llnb[i] = (const float*)d_in[b0 + 4];
    LW2[i]  = (const float*)d_in[b0 + 5];
    Lb2[i]  = (const float*)d_in[b0 + 6];
    Lng[i]  = (const float*)d_in[b0 + 7];
    Lnb[i]  = (const float*)d_in[b0 + 8];
  }
  const float* Norm_g = (const float*)d_in[40];
  const float* Norm_b = (const float*)d_in[41];
  const float* lin_W  = (const float*)d_in[42];
  const float* lin_b  = (const float*)d_in[43];

  // workspace layout (floats): H[N*512] | A[N*32] | m_enc[N*256] | denom[N*256] | agg[N*256] | T1[N*512]
  float*    H     = (float*)d_ws;
  float*    Aenc  = H + (long)N_NODES * 512;
  unsigned* m_enc = (unsigned*)(Aenc + (long)N_NODES * 32);
  float*    denom = (float*)m_enc + (long)N_NODES * 256;
  float*    agg   = denom + (long)N_NODES * 256;
  float*    T1    = agg + (long)N_NODES * 256;

  // 1) encoder: x @ enc_W + enc_b -> Aenc [N x 32]
  launch_gemm(x, NUM_FEATS, enc_W, 32, enc_b, Aenc, 32, N_NODES, NUM_FEATS, 32, stream);

  // 2) pre-loop conv (layer-0 weights), output straight into H[:,0:32]
  run_gen_conv(Aenc, 32, 32, 5, Lt[0], LW1[0], Lb1[0], Llng[0], Llnb[0], LW2[0], Lb2[0],
               H, 512, src, dst, m_enc, denom, agg, T1, stream);

  // 3) dense DeepGCN layers: conv on H[:,0:d], LN+ReLU, write into H[:,d:2d]
  for (int i = 0; i < 4; ++i) {
    int d = 32 << i, dshift = 5 + i;
    float* craw = (float*)m_enc;  // reuse m_enc region for raw conv output [N x d]
    run_gen_conv(H, 512, d, dshift, Lt[i], LW1[i], Lb1[i], Llng[i], Llnb[i], LW2[i], Lb2[i],
                 craw, d, src, dst, m_enc, denom, agg, T1, stream);
    k_row_ln<<<N_NODES, 128, 0, stream>>>(craw, d, H + d, 512, Lng[i], Lnb[i], d, 1);
  }

  // 4) final LayerNorm + ReLU on H[N x 512] -> xx output region
  float* out    = (float*)d_out;
  float* xx_out = out + (long)N_NODES * NUM_CLS;
  k_row_ln<<<N_NODES, 128, 0, stream>>>(H, 512, xx_out, 512, Norm_g, Norm_b, 512, 1);

  // 5) logits: xx @ lin_W + lin_b -> out[N x 7]
  int tl = N_NODES * NUM_CLS;
  k_logits<<<(tl + 255) / 256, 256, 0, stream>>>(xx_out, lin_W, lin_b, out);
}